// AeroChronoMixerForecaster_16664473109151
// MI455X (gfx1250) — compile-verified
//
#include <hip/hip_runtime.h>
#include <stdint.h>

// Instant-NGP multires hash encoding for MI455X (gfx1250).
//
// Bottleneck: 128M random 8B gathers into 16MB of tables (L2-resident in 192MB
// L2); HBM floor ~170MB (~7us @ 23.3TB/s). No GEMM structure -> WMMA N/A.
// CDNA5 path: per-lane async gathers to LDS (ASYNCcnt) in GVS/saddr form,
// double-buffered across levels with split s_wait_asynccnt — in-flight gather
// data costs no VGPRs and next level's loads overlap this level's interp.
// Hash math done directly in byte offsets (premultiplied primes, mask-before-
// xor) so each corner costs ~3 int VALU and zero 64-bit address arithmetic.
// NOTE: no min-waves launch bound — forcing 8 waves/SIMD made the allocator
// spill to scratch (vmem 70->126), which costs more than the occupancy gained.

#define NUM_LEVELS 8
#define TABLE_SIZE (1 << 18)
#define TBL_MASK (TABLE_SIZE - 1)
#define HP1 1540863u
#define HP2 1256879u
#define HP3 1957123u
#define THREADS 256

typedef float v2f __attribute__((ext_vector_type(2)));
typedef float v4f __attribute__((ext_vector_type(4)));

// Per-level prep: compute fractional coords (kept for the consume stage) and
// issue the 8 corner gathers as async global->LDS b64 loads in saddr form.
__device__ __forceinline__ void prep_issue(
    unsigned long long tbl_base,   // this level's table base address (SGPR pair)
    float xx, float yy, float zz, float res,
    const unsigned* lds_slot,      // 8 per-corner LDS addresses (target buffer)
    float& fx, float& fy, float& fz)
{
  const unsigned P18 = 8u * HP1, P28 = 8u * HP2, P38 = 8u * HP3;
  const unsigned M8 = (unsigned)TBL_MASK << 3;  // 0x1FFFF8: byte-offset mask
  const float px = xx * res, py = yy * res, pz = zz * res;
  const float bx = floorf(px), by = floorf(py), bz = floorf(pz);
  fx = px - bx; fy = py - by; fz = pz - bz;
  // corner byte offset = ((cx*P1 ^ cy*P2 ^ cz*P3) & TBL_MASK) * 8
  //                    = (cx*(8P1) ^ cy*(8P2) ^ cz*(8P3)) & (TBL_MASK<<3)
  // (low 21 bits exact under mod-2^32; mask distributes over XOR)
  const unsigned ix = (unsigned)(int)bx, iy = (unsigned)(int)by, iz = (unsigned)(int)bz;
  const unsigned a0 = ix * P18, b0 = iy * P28, c0 = iz * P38;
  const unsigned A0 = a0 & M8, A1 = (a0 + P18) & M8;
  const unsigned B0 = b0 & M8, B1 = (b0 + P28) & M8;
  const unsigned C0 = c0 & M8, C1 = (c0 + P38) & M8;
  const unsigned m00 = A0 ^ B0, m01 = A0 ^ B1, m10 = A1 ^ B0, m11 = A1 ^ B1;
  // k bits: [2]=x offset, [1]=y, [0]=z  (matches reference OFFSETS order)
  const unsigned off[8] = {m00 ^ C0, m00 ^ C1, m01 ^ C0, m01 ^ C1,
                           m10 ^ C0, m10 ^ C1, m11 ^ C0, m11 ^ C1};
#pragma unroll
  for (int k = 0; k < 8; ++k) {
    // CDNA5 async gather, GVS mode: SGPR64 base + VGPR32 byte offset -> LDS.
    asm volatile("global_load_async_to_lds_b64 %0, %1, %2"
                 :: "v"(lds_slot[k]), "v"(off[k]), "s"(tbl_base) : "memory");
  }
}

__global__ __launch_bounds__(THREADS) void hashgrid_encode_kernel(
    const float* __restrict__ x01,     // [N,3]
    const float* __restrict__ tables,  // [8, 2^18, 2]
    float* __restrict__ out,           // [N,16]
    int n_points)
{
  // [buffer][corner][tid] float2: conflict-free ds_load_b64 (lane t -> banks 2t,2t+1)
  __shared__ float2 sh[2][8][THREADS];  // 32 KB
  const int tid = threadIdx.x;
  const int n = blockIdx.x * THREADS + tid;
  const bool active = (n < n_points);
  const int nc = active ? n : (n_points - 1);  // tail lanes clamp; EXEC stays full

  const float* xp = x01 + 3ull * (unsigned)nc;
  const float xx = fminf(fmaxf(xp[0], 0.f), 1.f);
  const float yy = fminf(fmaxf(xp[1], 0.f), 1.f);
  const float zz = fminf(fmaxf(xp[2], 0.f), 1.f);

  // LDS byte address of this thread's slot (generic ptr low 32 = LDS offset).
  const unsigned lds0 =
      ((unsigned)(unsigned long long)(&sh[0][0][0])) + (unsigned)(tid * 8);
  const unsigned BUFSTRIDE = 8u * THREADS * 8u;   // bytes per buffer
  const unsigned CORNSTRIDE = (unsigned)THREADS * 8u;  // bytes per corner row

  // Precompute all 16 LDS slot addresses once (saves 8 v_adds per level).
  unsigned slot[2][8];
#pragma unroll
  for (int k = 0; k < 8; ++k) {
    slot[0][k] = lds0 + (unsigned)k * CORNSTRIDE;
    slot[1][k] = slot[0][k] + BUFSTRIDE;
  }

  // floor(16 * (256/16)^(l/7) + 1e-6), precomputed (all far from int boundaries)
  const int RES[NUM_LEVELS] = {16, 23, 35, 52, 78, 115, 172, 256};
  const unsigned long long tbl0 = (unsigned long long)(const void*)tables;
  const unsigned long long LVL_BYTES = (unsigned long long)TABLE_SIZE * 8ull;

  float fx_c, fy_c, fz_c, fx_n, fy_n, fz_n;
  // Prologue: level 0 gathers into buffer 0.
  prep_issue(tbl0, xx, yy, zz, (float)RES[0], slot[0], fx_c, fy_c, fz_c);

  v2f o2[NUM_LEVELS];
#pragma unroll
  for (int lvl = 0; lvl < NUM_LEVELS; ++lvl) {
    const int buf = lvl & 1;
    if (lvl + 1 < NUM_LEVELS) {
      // Prefetch next level into the other buffer, then wait only for the
      // current level's 8 loads (async loads complete in order).
      prep_issue(tbl0 + (unsigned long long)(lvl + 1) * LVL_BYTES,
                 xx, yy, zz, (float)RES[lvl + 1], slot[buf ^ 1],
                 fx_n, fy_n, fz_n);
      asm volatile("s_wait_asynccnt 0x8" ::: "memory");
    } else {
      asm volatile("s_wait_asynccnt 0x0" ::: "memory");
    }

    const float fx = fx_c, fy = fy_c, fz = fz_c;
    const float gx = 1.f - fx, gy = 1.f - fy, gz = 1.f - fz;

    // z-first trilinear blend in packed f32 (lowers to v_pk_fma_f32).
    const v2f e0 = *(const v2f*)&sh[buf][0][tid];
    const v2f e1 = *(const v2f*)&sh[buf][1][tid];
    const v2f e2 = *(const v2f*)&sh[buf][2][tid];
    const v2f e3 = *(const v2f*)&sh[buf][3][tid];
    const v2f e4 = *(const v2f*)&sh[buf][4][tid];
    const v2f e5 = *(const v2f*)&sh[buf][5][tid];
    const v2f e6 = *(const v2f*)&sh[buf][6][tid];
    const v2f e7 = *(const v2f*)&sh[buf][7][tid];
    const v2f t00 = e0 * gz + e1 * fz;   // x0 y0
    const v2f t01 = e2 * gz + e3 * fz;   // x0 y1
    const v2f t10 = e4 * gz + e5 * fz;   // x1 y0
    const v2f t11 = e6 * gz + e7 * fz;   // x1 y1
    const float w00 = gx * gy, w01 = gx * fy, w10 = fx * gy, w11 = fx * fy;
    o2[lvl] = t00 * w00 + t01 * w01 + t10 * w10 + t11 * w11;

    if (lvl + 1 < NUM_LEVELS) { fx_c = fx_n; fy_c = fy_n; fz_c = fz_n; }
  }

  if (active) {
    // 64B contiguous per point: 4x b128 stores, coalesced across the wave.
    // Nontemporal: 128MB streaming output must not evict the L2-resident tables.
    v4f* op = (v4f*)(out + 16ull * (unsigned)n);
    v4f v0 = {o2[0].x, o2[0].y, o2[1].x, o2[1].y};
    v4f v1 = {o2[2].x, o2[2].y, o2[3].x, o2[3].y};
    v4f v2 = {o2[4].x, o2[4].y, o2[5].x, o2[5].y};
    v4f v3 = {o2[6].x, o2[6].y, o2[7].x, o2[7].y};
    __builtin_nontemporal_store(v0, op + 0);
    __builtin_nontemporal_store(v1, op + 1);
    __builtin_nontemporal_store(v2, op + 2);
    __builtin_nontemporal_store(v3, op + 3);
  }
}

extern "C" void kernel_launch(void* const* d_in, const int* in_sizes, int n_in,
                              void* d_out, int out_size, void* d_ws, size_t ws_size,
                              hipStream_t stream) {
  const float* x01 = (const float*)d_in[0];     // [N,3] float32
  const float* tables = (const float*)d_in[1];  // [8, 2^18, 2] float32
  float* out = (float*)d_out;                   // [N,16] float32
  const int n = in_sizes[0] / 3;
  const int blocks = (n + THREADS - 1) / THREADS;
  hipLaunchKernelGGL(hashgrid_encode_kernel, dim3(blocks), dim3(THREADS), 0, stream,
                     x01, tables, out, n);
}